// SpatialTransformer_24842090840221
// MI455X (gfx1250) — compile-verified
//
#include <hip/hip_runtime.h>
#include <stdint.h>

// ---------------------------------------------------------------------------
// 3D spatial transformer (trilinear grid_sample, border padding,
// align_corners=False) for B=2, C=1, H=W=D=160.
//
// Coordinate simplification:  ix = clamp(d + (D/2)*dx, 0, D-1)  etc.
//
// MI455X strategy (memory-bound, ~164MB HBM traffic -> ~7us floor):
//  * deformation streamed via global_load_async_to_lds_b128 (ASYNCcnt path,
//    no VGPR writeback), consumed from LDS with ds_load_b128.
//  * moving (32.8MB total) is fully resident in the 192MB L2; the 8
//    gathers/voxel are L2 hits, not HBM traffic.
//  * non-temporal output stores so the write stream doesn't evict moving.
//  * VALU trimmed: launch geometry carries (b,h,w) so there is no %160//160
//    magic-multiply decode; neighbor gathers use one base address plus
//    select-of-constant offsets (dx/dy/dz in {0,1} voxels).
//  * wave32: each thread owns one 16B float4 chunk; it only reads back LDS
//    bytes its own async op wrote -> s_wait_asynccnt 0, no barrier.
// ---------------------------------------------------------------------------

typedef float v4f __attribute__((ext_vector_type(4)));

#define SD 160               // D (fastest spatial dim)
#define SW 160               // W
#define SH 160               // H
#define HWD (SH * SW * SD)   // 4,096,000
#define WROWS_PER_BLOCK 4    // 4 w-rows x 160 voxels = 640 voxels / block
#define BLOCK_THREADS 160    // 5 wave32; each thread = 4 voxels
#define VOX_PER_BLOCK (WROWS_PER_BLOCK * SD)  // 640

__global__ __launch_bounds__(BLOCK_THREADS)
void SpatialTransformer_24842090840221_kernel(const float* __restrict__ moving,
                                              const float* __restrict__ deform,
                                              float* __restrict__ out) {
    __shared__ __align__(16) float sdef[3][VOX_PER_BLOCK];  // 7.7 KB of 320 KB/WGP

    const int t = threadIdx.x;          // 0..159
    const int h = blockIdx.y;           // 0..159
    const int b = blockIdx.z;           // 0..1
    // this block's 640 contiguous voxels start here (scalar math):
    const int sbase = (h * SW + (int)blockIdx.x * WROWS_PER_BLOCK) * SD;
    const int s0 = sbase + t * 4;       // this thread's 4 voxels (16B aligned)

    // ---- async-stage this thread's 3 deformation components (16B each) ----
    const float* defb = deform + (size_t)b * 3 * HWD;
    {
        const uint64_t g0 = (uint64_t)(uintptr_t)(defb + (size_t)0 * HWD + s0);
        const uint64_t g1 = (uint64_t)(uintptr_t)(defb + (size_t)1 * HWD + s0);
        const uint64_t g2 = (uint64_t)(uintptr_t)(defb + (size_t)2 * HWD + s0);
        const uint32_t l0 = (uint32_t)(uintptr_t)(&sdef[0][t * 4]);
        const uint32_t l1 = (uint32_t)(uintptr_t)(&sdef[1][t * 4]);
        const uint32_t l2 = (uint32_t)(uintptr_t)(&sdef[2][t * 4]);
        asm volatile("global_load_async_to_lds_b128 %0, %1, off"
                     :: "v"(l0), "v"(g0) : "memory");
        asm volatile("global_load_async_to_lds_b128 %0, %1, off"
                     :: "v"(l1), "v"(g1) : "memory");
        asm volatile("global_load_async_to_lds_b128 %0, %1, off"
                     :: "v"(l2), "v"(g2) : "memory");
    }

    // decode (w, d) while the async DMA is in flight (only cheap t/40 here)
    const int wsub = t / 40;                        // 0..3 (w-row within block)
    const int d0   = (t - wsub * 40) * 4;           // 0,4,...,156
    const int w    = (int)blockIdx.x * WROWS_PER_BLOCK + wsub;
    const float wf = (float)w;
    const float hf = (float)h;
    const float* mv = moving + (size_t)b * HWD;

    asm volatile("s_wait_asynccnt 0" ::: "memory");

    const v4f dx4 = *reinterpret_cast<const v4f*>(&sdef[0][t * 4]);
    const v4f dy4 = *reinterpret_cast<const v4f*>(&sdef[1][t * 4]);
    const v4f dz4 = *reinterpret_cast<const v4f*>(&sdef[2][t * 4]);

    v4f res;
#pragma unroll
    for (int j = 0; j < 4; ++j) {
        // source coords: index + (n/2)*disp, clamped to the border
        const float fx = fminf(fmaxf((float)(d0 + j) + 80.0f * dx4[j], 0.0f), (float)(SD - 1));
        const float fy = fminf(fmaxf(wf            + 80.0f * dy4[j], 0.0f), (float)(SW - 1));
        const float fz = fminf(fmaxf(hf            + 80.0f * dz4[j], 0.0f), (float)(SH - 1));

        const int x0 = (int)fx;                 // floor (fx >= 0)
        const int y0 = (int)fy;
        const int z0 = (int)fz;
        const float wx = fx - (float)x0;
        const float wy = fy - (float)y0;
        const float wz = fz - (float)z0;
        // border clamp: neighbor offset collapses to 0 at the far edge
        const int xo = (x0 < SD - 1) ? 1        : 0;
        const int yo = (y0 < SW - 1) ? SD       : 0;
        const int zo = (z0 < SH - 1) ? SW * SD  : 0;

        const int base = (z0 * SW + y0) * SD + x0;
        const float c000 = mv[base],           c001 = mv[base + xo];
        const float c010 = mv[base + yo],      c011 = mv[base + yo + xo];
        const float c100 = mv[base + zo],      c101 = mv[base + zo + xo];
        const float c110 = mv[base + zo + yo], c111 = mv[base + zo + yo + xo];

        const float c00 = c000 + (c001 - c000) * wx;
        const float c01 = c010 + (c011 - c010) * wx;
        const float c10 = c100 + (c101 - c100) * wx;
        const float c11 = c110 + (c111 - c110) * wx;
        const float c0  = c00  + (c01  - c00 ) * wy;
        const float c1  = c10  + (c11  - c10 ) * wy;
        res[j] = c0 + (c1 - c0) * wz;
    }

    // non-temporal 16B store: don't let the output stream evict `moving` in L2
    __builtin_nontemporal_store(
        res, reinterpret_cast<v4f*>(out + (size_t)b * HWD + s0));
}

extern "C" void kernel_launch(void* const* d_in, const int* in_sizes, int n_in,
                              void* d_out, int out_size, void* d_ws, size_t ws_size,
                              hipStream_t stream) {
    (void)in_sizes; (void)n_in; (void)d_ws; (void)ws_size; (void)out_size;
    const float* moving = (const float*)d_in[0];   // [2,1,160,160,160] f32
    const float* deform = (const float*)d_in[1];   // [2,3,160,160,160] f32
    float* out = (float*)d_out;                    // [2,1,160,160,160] f32

    dim3 grid(SW / WROWS_PER_BLOCK, SH, 2);        // (40, 160, 2) = 12800 blocks
    SpatialTransformer_24842090840221_kernel<<<grid, BLOCK_THREADS, 0, stream>>>(
        moving, deform, out);
}